// SO3RotInvTransformer_79783312490941
// MI455X (gfx1250) — compile-verified
//
#include <hip/hip_runtime.h>
#include <hip/hip_bf16.h>
#include <math.h>

// ---------------------------------------------------------------------------
// SO3RotInvTransformer forward, MI455X (gfx1250), wave32, f32 WMMA (16x16x4).
// Dead-code-eliminated: the pairwise cdist / rel-dir MLP / masked softmax in
// the reference is discarded (`_unused`), so only the real attention path is
// computed. The two stacked input projections are folded into one effective
// 128x128 weight per q/k/v stream on-device.
// ---------------------------------------------------------------------------

#define DM   128      // d_model
#define SEQ  1024     // sequence length per batch
#define NB   4        // batches
#define BN   (NB*SEQ) // 4096 total rows
#define SROW 1025     // LDS score row stride (floats) -> conflict-free column walks
#define TROW 132      // LDS tile row stride (floats)

typedef __attribute__((ext_vector_type(2))) float v2f;
typedef __attribute__((ext_vector_type(8))) float v8f;

static __device__ __forceinline__ v8f wmma4(v2f a, v2f b, v8f c) {
  // V_WMMA_F32_16X16X4_F32 : D = A(16x4,f32) * B(4x16,f32) + C(16x16,f32)
  return __builtin_amdgcn_wmma_f32_16x16x4_f32(
      /*neg_a=*/false, a, /*neg_b=*/false, b,
      /*c_mod=*/(short)0, c, /*reuse_a=*/false, /*reuse_b=*/false);
}

// ---------------------------------------------------------------------------
// Kernel 1: fold (Wq,bq)+(wq2,bq2) -> (Aeff, ceff), likewise for k and v.
//   Aeff_p[m][n] = sum_o in_w[p*128+m][o] * W_p[o][n]
//   ceff_p[m]    = sum_o in_w[p*128+m][o] * b_p[o] + in_b[p*128+m]
// ---------------------------------------------------------------------------
__global__ __launch_bounds__(256) void fuse_weights_kernel(
    const float* __restrict__ Wq, const float* __restrict__ bq,
    const float* __restrict__ Wk, const float* __restrict__ bk,
    const float* __restrict__ Wv, const float* __restrict__ bv,
    const float* __restrict__ in_w, const float* __restrict__ in_b,
    float* __restrict__ Aeff, float* __restrict__ ceff) {
  const int p = blockIdx.x;                   // 0=q 1=k 2=v
  const float* W  = (p == 0) ? Wq : (p == 1) ? Wk : Wv;
  const float* bb = (p == 0) ? bq : (p == 1) ? bk : bv;
  const float* w2 = in_w + p * DM * DM;
  const float* b2 = in_b + p * DM;
  float* A = Aeff + p * DM * DM;
  float* c = ceff + p * DM;

  const int tid  = threadIdx.x;
  const int lane = tid & 31, wave = tid >> 5;
  const int mrow = lane & 15;                 // A-frag row / B-frag col / D col
  const int kh   = (lane >> 4) << 1;          // K half: lanes 16-31 hold K=2,3
  const int roff = (lane >> 4) << 3;          // D rows: lanes 16-31 hold M+8

  for (int t = 0; t < 8; ++t) {               // 8 waves x 8 tiles = 64 tiles
    const int tileid = wave * 8 + t;
    const int tm = (tileid >> 3) * 16, tn = (tileid & 7) * 16;
    v8f acc = {};
    for (int k0 = 0; k0 < DM; k0 += 4) {
      v2f a, b;
      a.x = w2[(tm + mrow) * DM + k0 + kh];
      a.y = w2[(tm + mrow) * DM + k0 + kh + 1];
      b.x = W[(k0 + kh) * DM + tn + mrow];
      b.y = W[(k0 + kh + 1) * DM + tn + mrow];
      acc = wmma4(a, b, acc);
    }
#pragma unroll
    for (int r = 0; r < 8; ++r)
      A[(tm + r + roff) * DM + tn + mrow] = acc[r];
  }
  if (tid < DM) {                             // folded bias
    float s = b2[tid];
    for (int o = 0; o < DM; ++o) s += w2[tid * DM + o] * bb[o];
    c[tid] = s;
  }
}

// ---------------------------------------------------------------------------
// Kernel 2: Q2/K2/V2[p] = src @ Aeff_p^T + ceff_p   (M=4096, N=128, K=128)
// grid (256 row-tiles, 3 streams), 8 waves: one 16x16 tile per wave.
// ---------------------------------------------------------------------------
__global__ __launch_bounds__(256) void proj_kernel(
    const float* __restrict__ src, const float* __restrict__ Aeff,
    const float* __restrict__ ceff, float* __restrict__ QKV) {
  const int p = blockIdx.y;
  const float* A = Aeff + p * DM * DM;
  const float* c = ceff + p * DM;
  float* out = QKV + (size_t)p * BN * DM;

  const int mb   = blockIdx.x * 16;
  const int tid  = threadIdx.x;
  const int lane = tid & 31, wave = tid >> 5;
  const int mrow = lane & 15;
  const int kh   = (lane >> 4) << 1;
  const int roff = (lane >> 4) << 3;
  const int nb   = wave * 16;

  v8f acc = {};
  for (int k0 = 0; k0 < DM; k0 += 4) {
    v2f a, b;
    a.x = src[(mb + mrow) * DM + k0 + kh];
    a.y = src[(mb + mrow) * DM + k0 + kh + 1];
    b.x = A[(nb + mrow) * DM + k0 + kh];      // B[k][n] = Aeff[n][k]
    b.y = A[(nb + mrow) * DM + k0 + kh + 1];
    acc = wmma4(a, b, acc);
  }
#pragma unroll
  for (int r = 0; r < 8; ++r)
    out[(size_t)(mb + r + roff) * DM + nb + mrow] = acc[r] + c[nb + mrow];
}

// ---------------------------------------------------------------------------
// Kernel 3: fused attention tail per 16-row tile:
//   S = Q2 K2^T / sqrt(D)  (16x1024, LDS)  -> softmax (1/sum deferred)
//   T1 = softmax(S) V2                      (16x128)
//   T2 = T1 @ out_w^T + out_b + src         (residual)
//   T3 = T2 @ Wd^T + bd  -> LayerNorm -> exact GELU -> d_out
// 4 waves / 128 threads; ~82 KB dynamic LDS (WGP has 320 KB).
// ---------------------------------------------------------------------------
__global__ __launch_bounds__(128) void attn_kernel(
    const float* __restrict__ src, const float* __restrict__ QKV,
    const float* __restrict__ out_w, const float* __restrict__ out_b,
    const float* __restrict__ Wd, const float* __restrict__ bd,
    const float* __restrict__ ln_g, const float* __restrict__ ln_b,
    float* __restrict__ out) {
  extern __shared__ float lds[];
  float* sc   = lds;                       // 16 * SROW  scores / exp(scores)
  float* t1   = sc + 16 * SROW;            // 16 * TROW
  float* t2   = t1 + 16 * TROW;            // 16 * TROW
  float* invs = t2 + 16 * TROW;            // 16

  const float* Q2 = QKV;
  const float* K2 = QKV + (size_t)BN * DM;
  const float* V2 = QKV + (size_t)2 * BN * DM;

  const int rt = blockIdx.x;               // 0..255
  const int b  = rt >> 6;
  const int gm = b * SEQ + (rt & 63) * 16; // global row base of this tile
  const int gb = b * SEQ;                  // key/value base row of this batch

  const int tid  = threadIdx.x;
  const int lane = tid & 31, wave = tid >> 5;
  const int mrow = lane & 15;
  const int kh   = (lane >> 4) << 1;
  const int roff = (lane >> 4) << 3;
  const float scale = 0.08838834764831845f; // 1/sqrt(128)

  // --- Phase 1: scores into LDS -------------------------------------------
  for (int ct = wave; ct < SEQ / 16; ct += 4) {
    const int nb = ct * 16;
    v8f acc = {};
    for (int k0 = 0; k0 < DM; k0 += 4) {
      v2f a, bf;
      a.x  = Q2[(size_t)(gm + mrow) * DM + k0 + kh];
      a.y  = Q2[(size_t)(gm + mrow) * DM + k0 + kh + 1];
      bf.x = K2[(size_t)(gb + nb + mrow) * DM + k0 + kh];   // B[k][n]=K2[n][k]
      bf.y = K2[(size_t)(gb + nb + mrow) * DM + k0 + kh + 1];
      acc = wmma4(a, bf, acc);
    }
#pragma unroll
    for (int r = 0; r < 8; ++r)
      sc[(r + roff) * SROW + nb + mrow] = acc[r] * scale;
  }
  __syncthreads();

  // --- Phase 2: row max, exp, sum (8 lanes per row) -----------------------
  {
    const int r = tid >> 3, sub = tid & 7;
    float mx = -3.402823466e38f;
    for (int j = sub; j < SEQ; j += 8) mx = fmaxf(mx, sc[r * SROW + j]);
    mx = fmaxf(mx, __shfl_xor(mx, 1));
    mx = fmaxf(mx, __shfl_xor(mx, 2));
    mx = fmaxf(mx, __shfl_xor(mx, 4));
    float sum = 0.f;
    for (int j = sub; j < SEQ; j += 8) {
      float e = expf(sc[r * SROW + j] - mx);
      sc[r * SROW + j] = e;
      sum += e;
    }
    sum += __shfl_xor(sum, 1);
    sum += __shfl_xor(sum, 2);
    sum += __shfl_xor(sum, 4);
    if (sub == 0) invs[r] = 1.0f / sum;
  }
  __syncthreads();

  // --- Phase 3: T1 = exp(S) @ V2, rows scaled by 1/sum --------------------
  for (int ct = wave * 2; ct < wave * 2 + 2; ++ct) {
    const int nb = ct * 16;
    v8f acc = {};
    for (int k0 = 0; k0 < SEQ; k0 += 4) {
      v2f a, bf;
      a.x  = sc[mrow * SROW + k0 + kh];
      a.y  = sc[mrow * SROW + k0 + kh + 1];
      bf.x = V2[(size_t)(gb + k0 + kh) * DM + nb + mrow];
      bf.y = V2[(size_t)(gb + k0 + kh + 1) * DM + nb + mrow];
      acc = wmma4(a, bf, acc);
    }
#pragma unroll
    for (int r = 0; r < 8; ++r) {
      const int row = r + roff;
      t1[row * TROW + nb + mrow] = acc[r] * invs[row];
    }
  }
  __syncthreads();

  // --- Phase 4: T2 = T1 @ out_w^T + out_b + src (residual) ----------------
  for (int ct = wave * 2; ct < wave * 2 + 2; ++ct) {
    const int nb = ct * 16;
    v8f acc = {};
    for (int k0 = 0; k0 < DM; k0 += 4) {
      v2f a, bf;
      a.x  = t1[mrow * TROW + k0 + kh];
      a.y  = t1[mrow * TROW + k0 + kh + 1];
      bf.x = out_w[(nb + mrow) * DM + k0 + kh];
      bf.y = out_w[(nb + mrow) * DM + k0 + kh + 1];
      acc = wmma4(a, bf, acc);
    }
#pragma unroll
    for (int r = 0; r < 8; ++r) {
      const int row = r + roff, col = nb + mrow;
      t2[row * TROW + col] =
          acc[r] + out_b[col] + src[(size_t)(gm + row) * DM + col];
    }
  }
  __syncthreads();

  // --- Phase 5: T3 = T2 @ Wd^T + bd  (into t1) ----------------------------
  for (int ct = wave * 2; ct < wave * 2 + 2; ++ct) {
    const int nb = ct * 16;
    v8f acc = {};
    for (int k0 = 0; k0 < DM; k0 += 4) {
      v2f a, bf;
      a.x  = t2[mrow * TROW + k0 + kh];
      a.y  = t2[mrow * TROW + k0 + kh + 1];
      bf.x = Wd[(nb + mrow) * DM + k0 + kh];
      bf.y = Wd[(nb + mrow) * DM + k0 + kh + 1];
      acc = wmma4(a, bf, acc);
    }
#pragma unroll
    for (int r = 0; r < 8; ++r)
      t1[(r + roff) * TROW + nb + mrow] = acc[r] + bd[nb + mrow];
  }
  __syncthreads();

  // --- Phase 6: LayerNorm + exact GELU + store ----------------------------
  {
    const int r = tid >> 3, sub = tid & 7;
    float s1 = 0.f;
    for (int j = sub; j < DM; j += 8) s1 += t1[r * TROW + j];
    s1 += __shfl_xor(s1, 1);
    s1 += __shfl_xor(s1, 2);
    s1 += __shfl_xor(s1, 4);
    const float mu = s1 * (1.0f / DM);
    float s2 = 0.f;
    for (int j = sub; j < DM; j += 8) {
      const float d = t1[r * TROW + j] - mu;
      s2 += d * d;
    }
    s2 += __shfl_xor(s2, 1);
    s2 += __shfl_xor(s2, 2);
    s2 += __shfl_xor(s2, 4);
    const float rstd = 1.0f / sqrtf(s2 * (1.0f / DM) + 1e-5f);
    for (int j = sub; j < DM; j += 8) {
      const float x = (t1[r * TROW + j] - mu) * rstd * ln_g[j] + ln_b[j];
      out[(size_t)(gm + r) * DM + j] =
          0.5f * x * (1.0f + erff(x * 0.7071067811865475f));
    }
  }
}

// ---------------------------------------------------------------------------
extern "C" void kernel_launch(void* const* d_in, const int* in_sizes, int n_in,
                              void* d_out, int out_size, void* d_ws,
                              size_t ws_size, hipStream_t stream) {
  const float* src   = (const float*)d_in[0];
  // d_in[1]=ray_3d, d_in[2]=ray_dir : dead (discarded softmax branch)
  const float* Wq    = (const float*)d_in[3];
  const float* bq    = (const float*)d_in[4];
  const float* Wk    = (const float*)d_in[5];
  const float* bk    = (const float*)d_in[6];
  const float* Wv    = (const float*)d_in[7];
  const float* bv    = (const float*)d_in[8];
  // d_in[9..12] = W1,b1,W2,b2 : dead (rel_dir MLP feeds the discarded branch)
  const float* in_w  = (const float*)d_in[13];
  const float* in_b  = (const float*)d_in[14];
  const float* out_w = (const float*)d_in[15];
  const float* out_b = (const float*)d_in[16];
  const float* Wd    = (const float*)d_in[17];
  const float* bd    = (const float*)d_in[18];
  const float* ln_g  = (const float*)d_in[19];
  const float* ln_b  = (const float*)d_in[20];

  float* ws   = (float*)d_ws;
  float* Aeff = ws;                         // 3 * 128*128
  float* ceff = Aeff + 3 * DM * DM;         // 3 * 128
  float* QKV  = ceff + 3 * DM;              // 3 * 4096*128  (Q2,K2,V2)

  fuse_weights_kernel<<<3, 256, 0, stream>>>(Wq, bq, Wk, bk, Wv, bv, in_w,
                                             in_b, Aeff, ceff);
  proj_kernel<<<dim3(BN / 16, 3), 256, 0, stream>>>(src, Aeff, ceff, QKV);

  const size_t lds_bytes =
      (size_t)(16 * SROW + 2 * 16 * TROW + 16) * sizeof(float);
  attn_kernel<<<BN / 16, 128, lds_bytes, stream>>>(
      src, QKV, out_w, out_b, Wd, bd, ln_g, ln_b, (float*)d_out);
}